// GATv2ConvThr_33191507263710
// MI455X (gfx1250) — compile-verified
//
#include <hip/hip_runtime.h>
#include <math.h>

#define N_NODES 10000
#define M_EDGES 320000
#define F_IN    256
#define HEADS   4
#define HC      256          // HEADS * C
#define THR_A   1e-4f
#define THR_W   1e-2f
#define NEG_SLOPE 0.2f

typedef __attribute__((ext_vector_type(2))) float v2f;
typedef __attribute__((ext_vector_type(8))) float v8f;

// ---- workspace layout (float offsets, all 16B-aligned where needed) ----
#define OFF_COLSUMSQ 0                          // 256
#define OFF_THRWI    256                        // 256
#define OFF_SUMMSG   512                        // 1 (pad to 768)
#define OFF_DENOM    768                        // N*4 = 40000
#define OFF_AMAX     (OFF_DENOM + 4*N_NODES)    // 40768, N*4 uints
#define OFF_WLP      (OFF_AMAX + 4*N_NODES)     // 80768, 65536
#define OFF_WRP      (OFF_WLP + 65536)          // 146304
#define OFF_XL       (OFF_WRP + 65536)          // 211840, N*256
#define OFF_XR       (OFF_XL + N_NODES*HC)      // + 2,560,000
#define OFF_ALPHA    (OFF_XR + N_NODES*HC)      // M*4
#define OFF_MSGNORM  (OFF_ALPHA + 4*M_EDGES)    // M
// total ~6.93M floats (~27.7 MB)

__global__ void zero_kernel(float* __restrict__ p, int n) {
    int t = blockIdx.x * blockDim.x + threadIdx.x;
    if (t < n) p[t] = 0.0f;
}

// ---- per-column sum of squares of x (coalesced: thread = column) ----
__global__ void colnorm_kernel(const float* __restrict__ x, float* __restrict__ colsumsq) {
    int t = threadIdx.x;                 // column 0..255
    int r0 = blockIdx.x * 250;           // 40 blocks * 250 rows = 10000
    float s = 0.f;
    for (int r = 0; r < 250; ++r) {
        float v = x[(r0 + r) * F_IN + t];
        s += v * v;
    }
    atomicAdd(&colsumsq[t], s);
}

// ---- thr_wi[f] = THR_W * (||all||/256) / ||col f|| ----
__global__ void thr_kernel(const float* __restrict__ colsumsq, float* __restrict__ thrwi) {
    __shared__ float red[256];
    int t = threadIdx.x;
    float v = colsumsq[t];
    red[t] = v;
    __syncthreads();
    for (int s = 128; s > 0; s >>= 1) {
        if (t < s) red[t] += red[t + s];
        __syncthreads();
    }
    float normAll = sqrtf(red[0]) * (1.0f / 256.0f);
    thrwi[t] = THR_W * normAll / sqrtf(v);
}

// ---- prune W against thr_wi and pack into WMMA B-fragment order ----
// packed idx = ((ct*64 + ks)*32 + lane)*2 + r   holds  W[ct*16 + lane%16, ks*4 + r + 2*(lane/16)]
__global__ void pack_kernel(const float* __restrict__ Wl, const float* __restrict__ Wr,
                            const float* __restrict__ thrwi,
                            float* __restrict__ WlP, float* __restrict__ WrP) {
    int t = blockIdx.x * 256 + threadIdx.x;       // 2 * 65536 threads
    int which = t >> 16;
    int idx = t & 65535;
    int r  = idx & 1;
    int l  = (idx >> 1) & 31;
    int ks = (idx >> 6) & 63;
    int ct = idx >> 12;
    int o = ct * 16 + (l & 15);
    int f = ks * 4 + r + 2 * (l >> 4);
    const float* W = which ? Wr : Wl;
    float w = W[o * F_IN + f];
    float v = (fabsf(w) >= thrwi[f]) ? w : 0.0f;
    (which ? WrP : WlP)[idx] = v;
}

// ---- GEMM: x[strip 16 rows, 256] @ {WlP,WrP}^T -> x_l/x_r, via v_wmma_f32_16x16x4_f32 ----
#define TILE_PITCH 260   // 4-bank stride per row -> conflict-free ds_load_b64 A-frags
__global__ void gemm_kernel(const float* __restrict__ x,
                            const float* __restrict__ WlP, const float* __restrict__ WrP,
                            float* __restrict__ xl, float* __restrict__ xr) {
    __shared__ float tile[16 * TILE_PITCH];
    int strip = blockIdx.x;                       // 0..624
    int t = threadIdx.x;
    // stage 16x256 x-tile into LDS, coalesced float4
    const float* xbase = x + strip * 16 * F_IN;
    for (int i = t; i < 1024; i += 256) {         // 1024 float4 = 4096 floats
        int r = i >> 6, c4 = i & 63;
        ((float4*)(tile + r * TILE_PITCH))[c4] = ((const float4*)(xbase + r * F_IN))[c4];
    }
    __syncthreads();

    int wave = t >> 5, lane = t & 31;
    int m    = lane & 15;
    int koff = (lane >> 4) * 2;
    int mat  = wave >> 2;                         // 0: Wl, 1: Wr
    const float* Wp = mat ? WrP : WlP;
    float* out = mat ? xr : xl;

    for (int ti = 0; ti < 4; ++ti) {
        int ct = (wave & 3) * 4 + ti;             // col tile 0..15
        v8f c = {};
        const v2f* bptr = ((const v2f*)Wp) + ct * 64 * 32 + lane;
        #pragma unroll 4
        for (int ks = 0; ks < 64; ++ks) {
            v2f a = *(const v2f*)(&tile[m * TILE_PITCH + ks * 4 + koff]);
            v2f b = bptr[ks * 32];
            c = __builtin_amdgcn_wmma_f32_16x16x4_f32(
                    false, a, false, b, (short)0, c, false, false);
        }
        int col = ct * 16 + m;
        int rowbase = strip * 16 + 8 * (lane >> 4);
        #pragma unroll
        for (int v = 0; v < 8; ++v)
            out[(rowbase + v) * HC + col] = c[v];
    }
}

// ---- pass A: raw attention logits, segment max (uint-key atomicMax), msg norms ----
__global__ void edgeA_kernel(const float* __restrict__ xl, const float* __restrict__ xr,
                             const int* __restrict__ ei, const float* __restrict__ att,
                             float* __restrict__ alphaEx, float* __restrict__ msgnorm,
                             unsigned int* __restrict__ amaxKeys, float* __restrict__ sumMsg) {
    int wave = threadIdx.x >> 5, lane = threadIdx.x & 31;
    int m = blockIdx.x * 8 + wave;
    int src = ei[m];
    int dst = ei[M_EDGES + m];
    int ch = lane * 8;                            // head = lane/8
    float aj[8], xi[8], at[8];
    *(float4*)&aj[0] = *(const float4*)(xl + src * HC + ch);
    *(float4*)&aj[4] = *(const float4*)(xl + src * HC + ch + 4);
    *(float4*)&xi[0] = *(const float4*)(xr + dst * HC + ch);
    *(float4*)&xi[4] = *(const float4*)(xr + dst * HC + ch + 4);
    *(float4*)&at[0] = *(const float4*)(att + ch);
    *(float4*)&at[4] = *(const float4*)(att + ch + 4);
    float s = 0.f, q = 0.f;
    #pragma unroll
    for (int j = 0; j < 8; ++j) {
        float e = aj[j] + xi[j];
        e = (e >= 0.f) ? e : NEG_SLOPE * e;
        s += e * at[j];
        q += aj[j] * aj[j];
    }
    s += __shfl_xor(s, 1, 32);  s += __shfl_xor(s, 2, 32);  s += __shfl_xor(s, 4, 32);
    q += __shfl_xor(q, 1, 32);  q += __shfl_xor(q, 2, 32);  q += __shfl_xor(q, 4, 32);
    q += __shfl_xor(q, 8, 32);  q += __shfl_xor(q, 16, 32);
    if (lane == 0) {
        float nrm = sqrtf(q);
        msgnorm[m] = nrm;
        atomicAdd(sumMsg, nrm);
    }
    if ((lane & 7) == 0) {
        int h = lane >> 3;
        alphaEx[m * 4 + h] = s;
        unsigned int bits = __float_as_uint(s);
        unsigned int key = (bits & 0x80000000u) ? ~bits : (bits | 0x80000000u);
        atomicMax(&amaxKeys[dst * 4 + h], key);
    }
}

// ---- pass B: exp(alpha - amax), segment-sum denom ----
__global__ void edgeB_kernel(const int* __restrict__ ei, float* __restrict__ alphaEx,
                             const unsigned int* __restrict__ amaxKeys,
                             float* __restrict__ denom) {
    int t = blockIdx.x * blockDim.x + threadIdx.x;
    if (t >= 4 * M_EDGES) return;
    int m = t >> 2, h = t & 3;
    int dst = ei[M_EDGES + m];
    unsigned int u = amaxKeys[dst * 4 + h];
    unsigned int bits = (u & 0x80000000u) ? (u & 0x7FFFFFFFu) : ~u;
    float amax = __uint_as_float(bits);
    float ex = __expf(alphaEx[t] - amax);
    alphaEx[t] = ex;
    atomicAdd(&denom[dst * 4 + h], ex);
}

// ---- pass C: softmax, threshold mask, scatter-add weighted messages ----
__global__ void edgeC_kernel(const float* __restrict__ xl, const int* __restrict__ ei,
                             const float* __restrict__ alphaEx, const float* __restrict__ denom,
                             const float* __restrict__ msgnorm, const float* __restrict__ sumMsg,
                             float* __restrict__ out) {
    int wave = threadIdx.x >> 5, lane = threadIdx.x & 31;
    int m = blockIdx.x * 8 + wave;
    int src = ei[m];
    int dst = ei[M_EDGES + m];
    float a = 0.f;
    if (lane < 4) a = alphaEx[m * 4 + lane] / denom[dst * 4 + lane];
    float s2 = a * a;
    s2 += __shfl_xor(s2, 1, 32);
    s2 += __shfl_xor(s2, 2, 32);
    float rownorm = sqrtf(__shfl(s2, 0, 32)) * (1.0f / HEADS);
    float normAll = sumMsg[0] * (1.0f / M_EDGES);
    float thr = THR_A * normAll / msgnorm[m];
    bool mask = (rownorm < thr) && (src != dst);   // wave-uniform
    if (mask) a = 0.f;
    float av = __shfl(a, lane >> 3, 32);           // broadcast this lane's head weight
    int ch = lane * 8;
    float xv[8];
    *(float4*)&xv[0] = *(const float4*)(xl + src * HC + ch);
    *(float4*)&xv[4] = *(const float4*)(xl + src * HC + ch + 4);
    float* op = out + dst * HC + ch;
    #pragma unroll
    for (int j = 0; j < 8; ++j) atomicAdd(op + j, xv[j] * av);
}

__global__ void bias_kernel(float* __restrict__ out, const float* __restrict__ bias, int n) {
    int t = blockIdx.x * blockDim.x + threadIdx.x;
    if (t < n) out[t] += bias[t & (HC - 1)];
}

extern "C" void kernel_launch(void* const* d_in, const int* in_sizes, int n_in,
                              void* d_out, int out_size, void* d_ws, size_t ws_size,
                              hipStream_t stream) {
    const float* x    = (const float*)d_in[0];
    const int*   ei   = (const int*)d_in[1];
    const float* Wl   = (const float*)d_in[2];
    const float* Wr   = (const float*)d_in[3];
    const float* att  = (const float*)d_in[4];
    const float* bias = (const float*)d_in[5];
    float* out = (float*)d_out;
    float* ws  = (float*)d_ws;

    float* colsumsq = ws + OFF_COLSUMSQ;
    float* thrwi    = ws + OFF_THRWI;
    float* sumMsg   = ws + OFF_SUMMSG;
    float* denom    = ws + OFF_DENOM;
    unsigned int* amaxKeys = (unsigned int*)(ws + OFF_AMAX);
    float* WlP  = ws + OFF_WLP;
    float* WrP  = ws + OFF_WRP;
    float* xl   = ws + OFF_XL;
    float* xr   = ws + OFF_XR;
    float* alphaEx = ws + OFF_ALPHA;
    float* msgnorm = ws + OFF_MSGNORM;

    // zero stats region (colsumsq..amaxKeys) and output accumulator
    int nstat = OFF_WLP;                                      // 80768 floats
    zero_kernel<<<(nstat + 255) / 256, 256, 0, stream>>>(ws, nstat);
    zero_kernel<<<(N_NODES * HC + 255) / 256, 256, 0, stream>>>(out, N_NODES * HC);

    colnorm_kernel<<<40, 256, 0, stream>>>(x, colsumsq);
    thr_kernel<<<1, 256, 0, stream>>>(colsumsq, thrwi);
    pack_kernel<<<512, 256, 0, stream>>>(Wl, Wr, thrwi, WlP, WrP);
    gemm_kernel<<<N_NODES / 16, 256, 0, stream>>>(x, WlP, WrP, xl, xr);
    edgeA_kernel<<<M_EDGES / 8, 256, 0, stream>>>(xl, xr, ei, att, alphaEx, msgnorm,
                                                  amaxKeys, sumMsg);
    edgeB_kernel<<<(4 * M_EDGES + 255) / 256, 256, 0, stream>>>(ei, alphaEx, amaxKeys, denom);
    edgeC_kernel<<<M_EDGES / 8, 256, 0, stream>>>(xl, ei, alphaEx, denom, msgnorm, sumMsg, out);
    bias_kernel<<<(N_NODES * HC + 255) / 256, 256, 0, stream>>>(out, bias, N_NODES * HC);
}